// GraphMeshConvolution_68547678044330
// MI455X (gfx1250) — compile-verified
//
#include <hip/hip_runtime.h>

// ---- CDNA5 WMMA types (wave32): f32 16x16x4 -> A,B = 2 VGPRs, C/D = 8 VGPRs
typedef float v2f __attribute__((ext_vector_type(2)));
typedef float v8f __attribute__((ext_vector_type(8)));

#define SLOPE 0.01f

__device__ __forceinline__ float leaky(float v) { return v > 0.f ? v : SLOPE * v; }
__device__ __forceinline__ float rsclamp(float d) { return rsqrtf(fmaxf(d, 1.f)); }

// ---------------------------------------------------------------- zero fill
__global__ void k_zero(float* __restrict__ p, long n4) {
    long i = (long)blockIdx.x * blockDim.x + threadIdx.x;
    long stride = (long)gridDim.x * blockDim.x;
    float4 z = make_float4(0.f, 0.f, 0.f, 0.f);
    for (; i < n4; i += stride) ((float4*)p)[i] = z;
}

// ---------------------------------------------------------------- degrees
__global__ void k_deg(const int* __restrict__ src, const int* __restrict__ dst,
                      float* __restrict__ outdeg, float* __restrict__ indeg, int E) {
    int e = blockIdx.x * blockDim.x + threadIdx.x;
    if (e < E) {
        atomicAdd(&outdeg[src[e]], 1.f);
        atomicAdd(&indeg[dst[e]], 1.f);
    }
}

// ---------------------------------------------------------------- GEMM1: h1[N,64] = (x * rs_out) @ W1[128,64]
// 256 threads = 8 waves; wave -> 16x16 output tile. 2 row tiles x 4 col tiles = 32 rows/block.
__global__ void k_gemm1(const float* __restrict__ x, const float* __restrict__ W1,
                        const float* __restrict__ outdeg, float* __restrict__ h1, int n) {
    const int lane = threadIdx.x & 31;
    const int wave = threadIdx.x >> 5;
    const int half = lane >> 4;   // 0: K pair {0,1}; 1: K pair {2,3}
    const int l16  = lane & 15;
    const int rowTile = blockIdx.x * 32 + (wave >> 2) * 16;
    const int colTile = (wave & 3) * 16;

    int arow = rowTile + l16;
    if (arow >= n) arow = n - 1;                 // clamp (keeps EXEC all-ones)
    const float rs = rsclamp(outdeg[arow]);
    const float* __restrict__ xrow = x + (long)arow * 128;

    v8f acc = {};
#pragma unroll
    for (int k = 0; k < 128; k += 4) {
        const int ka = k + half * 2;
        float2 av = *(const float2*)(xrow + ka);
        v2f a; a.x = av.x * rs; a.y = av.y * rs;
        v2f b;
        b.x = W1[(long)ka * 64 + colTile + l16];
        b.y = W1[(long)(ka + 1) * 64 + colTile + l16];
        acc = __builtin_amdgcn_wmma_f32_16x16x4_f32(false, a, false, b,
                                                    (short)0, acc, false, false);
    }
#pragma unroll
    for (int v = 0; v < 8; ++v) {
        int r = rowTile + half * 8 + v;
        if (r < n) h1[(long)r * 64 + colTile + l16] = acc[v];
    }
}

// ---------------------------------------------------------------- scatter layer 1: m1[dst] += h1[src]   (64 cols)
__global__ void k_scatter1(const int* __restrict__ src, const int* __restrict__ dst,
                           const float* __restrict__ h1, float* __restrict__ m1, int E) {
    long t = (long)blockIdx.x * blockDim.x + threadIdx.x;
    if (t >= (long)E * 16) return;
    int e = (int)(t >> 4);
    int c = (int)(t & 15) * 4;
    int s = src[e], d = dst[e];
    float4 v = *(const float4*)(h1 + (long)s * 64 + c);
    float* mp = m1 + (long)d * 64 + c;
    atomicAdd(mp + 0, v.x);
    atomicAdd(mp + 1, v.y);
    atomicAdd(mp + 2, v.z);
    atomicAdd(mp + 3, v.w);
}

// ---------------------------------------------------------------- GEMM2: h2[N,32] = (leaky(m1*rs_in)*rs_out) @ W2[64,32]
// 256 threads = 8 waves; 4 row tiles x 2 col tiles = 64 rows/block.
__global__ void k_gemm2(const float* __restrict__ m1, const float* __restrict__ W2,
                        const float* __restrict__ outdeg, const float* __restrict__ indeg,
                        float* __restrict__ h2, int n) {
    const int lane = threadIdx.x & 31;
    const int wave = threadIdx.x >> 5;
    const int half = lane >> 4;
    const int l16  = lane & 15;
    const int rowTile = blockIdx.x * 64 + (wave >> 1) * 16;
    const int colTile = (wave & 1) * 16;

    int arow = rowTile + l16;
    if (arow >= n) arow = n - 1;
    const float rsin  = rsclamp(indeg[arow]);
    const float rsout = rsclamp(outdeg[arow]);
    const float* __restrict__ mrow = m1 + (long)arow * 64;

    v8f acc = {};
#pragma unroll
    for (int k = 0; k < 64; k += 4) {
        const int ka = k + half * 2;
        float2 av = *(const float2*)(mrow + ka);
        v2f a;
        a.x = leaky(av.x * rsin) * rsout;
        a.y = leaky(av.y * rsin) * rsout;
        v2f b;
        b.x = W2[(long)ka * 32 + colTile + l16];
        b.y = W2[(long)(ka + 1) * 32 + colTile + l16];
        acc = __builtin_amdgcn_wmma_f32_16x16x4_f32(false, a, false, b,
                                                    (short)0, acc, false, false);
    }
#pragma unroll
    for (int v = 0; v < 8; ++v) {
        int r = rowTile + half * 8 + v;
        if (r < n) h2[(long)r * 32 + colTile + l16] = acc[v];
    }
}

// ---------------------------------------------------------------- scatter layer 2: m2[dst] += h2[src]   (32 cols)
__global__ void k_scatter2(const int* __restrict__ src, const int* __restrict__ dst,
                           const float* __restrict__ h2, float* __restrict__ m2, int E) {
    long t = (long)blockIdx.x * blockDim.x + threadIdx.x;
    if (t >= (long)E * 8) return;
    int e = (int)(t >> 3);
    int c = (int)(t & 7) * 4;
    int s = src[e], d = dst[e];
    float4 v = *(const float4*)(h2 + (long)s * 32 + c);
    float* mp = m2 + (long)d * 32 + c;
    atomicAdd(mp + 0, v.x);
    atomicAdd(mp + 1, v.y);
    atomicAdd(mp + 2, v.z);
    atomicAdd(mp + 3, v.w);
}

// ---------------------------------------------------------------- pooling: pooled[g] += leaky(m2*rs_in); cnt[g] += 1
__global__ void k_pool(const float* __restrict__ m2, const float* __restrict__ indeg,
                       const int* __restrict__ gid, float* __restrict__ pooled,
                       float* __restrict__ cnt, int n) {
    long t = (long)blockIdx.x * blockDim.x + threadIdx.x;
    if (t >= (long)n * 8) return;
    int node = (int)(t >> 3);
    int c = (int)(t & 7) * 4;
    float rsin = rsclamp(indeg[node]);
    int g = gid[node];
    float4 v = *(const float4*)(m2 + (long)node * 32 + c);
    float* pp = pooled + (long)g * 32 + c;
    atomicAdd(pp + 0, leaky(v.x * rsin));
    atomicAdd(pp + 1, leaky(v.y * rsin));
    atomicAdd(pp + 2, leaky(v.z * rsin));
    atomicAdd(pp + 3, leaky(v.w * rsin));
    if ((t & 7) == 0) atomicAdd(&cnt[g], 1.f);
}

// ---------------------------------------------------------------- head: out[G,16] = (pooled/cnt) @ Wc[32,16]
__global__ void k_head(const float* __restrict__ pooled, const float* __restrict__ cnt,
                       const float* __restrict__ Wc, float* __restrict__ out, int G) {
    int t = blockIdx.x * blockDim.x + threadIdx.x;
    if (t >= G * 16) return;
    int g = t >> 4, o = t & 15;
    float inv = 1.f / fmaxf(cnt[g], 1.f);
    float acc = 0.f;
#pragma unroll
    for (int k = 0; k < 32; ++k) acc += pooled[g * 32 + k] * inv * Wc[k * 16 + o];
    out[t] = acc;
}

// ================================================================ launch
extern "C" void kernel_launch(void* const* d_in, const int* in_sizes, int n_in,
                              void* d_out, int out_size, void* d_ws, size_t ws_size,
                              hipStream_t stream) {
    const float* x   = (const float*)d_in[0];
    const int*   src = (const int*)d_in[1];
    const int*   dst = (const int*)d_in[2];
    const int*   gid = (const int*)d_in[3];
    const float* W1  = (const float*)d_in[4];
    const float* W2  = (const float*)d_in[5];
    const float* Wc  = (const float*)d_in[6];
    float* out = (float*)d_out;

    const int  n = in_sizes[3];          // N nodes (graph_ids length)
    const int  E = in_sizes[1];          // edges
    const int  G = out_size / 16;        // graphs

    const long N = n;
    float* ws = (float*)d_ws;
    // layout (floats): [outdeg N][indeg N][m1 64N][m2 32N][pooled 32G][cnt G][h1 64N][h2 32N]
    float* outdeg = ws;
    float* indeg  = ws + N;
    float* m1     = ws + 2 * N;
    float* m2     = ws + 66 * N;
    float* pooled = ws + 98 * N;
    float* cnt    = pooled + (long)G * 32;
    long zlen = 98 * N + (long)G * 33;           // everything needing zero-init
    zlen = (zlen + 3) & ~3L;                     // float4 align
    float* h1 = ws + zlen;
    float* h2 = h1 + 64 * N;
    // total workspace: ~ (194*N + 33*G) * 4 bytes  (~78 MB for N=100000)

    // 1) zero accumulators
    {
        long n4 = zlen / 4;
        int blocks = (int)((n4 + 255) / 256);
        if (blocks > 4096) blocks = 4096;
        k_zero<<<blocks, 256, 0, stream>>>(ws, n4);
    }
    // 2) degrees
    k_deg<<<(E + 255) / 256, 256, 0, stream>>>(src, dst, outdeg, indeg, E);
    // 3) layer-1 projection (WMMA f32)
    k_gemm1<<<(n + 31) / 32, 256, 0, stream>>>(x, W1, outdeg, h1, n);
    // 4) layer-1 edge aggregation
    {
        long tot = (long)E * 16;
        k_scatter1<<<(int)((tot + 255) / 256), 256, 0, stream>>>(src, dst, h1, m1, E);
    }
    // 5) layer-2 projection (norm + leaky fused into A, WMMA f32)
    k_gemm2<<<(n + 63) / 64, 256, 0, stream>>>(m1, W2, outdeg, indeg, h2, n);
    // 6) layer-2 edge aggregation
    {
        long tot = (long)E * 8;
        k_scatter2<<<(int)((tot + 255) / 256), 256, 0, stream>>>(src, dst, h2, m2, E);
    }
    // 7) mean-pool accumulation
    {
        long tot = (long)n * 8;
        k_pool<<<(int)((tot + 255) / 256), 256, 0, stream>>>(m2, indeg, gid, pooled, cnt, n);
    }
    // 8) classifier head
    k_head<<<(G * 16 + 255) / 256, 256, 0, stream>>>(pooled, cnt, Wc, out, G);
}